// DeepStitch_51616916963778
// MI455X (gfx1250) — compile-verified
//
#include <hip/hip_runtime.h>
#include <hip/hip_bf16.h>

typedef __attribute__((ext_vector_type(2))) float v2f;
typedef __attribute__((ext_vector_type(8))) float v8f;

#define B_      4
#define COUT    256
#define OHW     128
#define NPIX    16384   // 128*128
#define K2      256     // 16*16 selected descriptors
#define HID     128

// ---------------------------------------------------------------------------
// Kernel 1: Conv2d(3->256,k3,s2,p1) + ReLU, fused per-pixel channel reduction.
// mode==0: red_out = sum_c f   (response map for adaptive max pool on fA)
// mode==1: red_out = sum_c f*f (||fB_n||^2 for the distance)
// grid = B*128*128 blocks (one output pixel), block = 256 threads (one per oc)
// ---------------------------------------------------------------------------
__global__ void __launch_bounds__(256)
conv_kernel(const float* __restrict__ x, const float* __restrict__ Wc,
            const float* __restrict__ bc, float* __restrict__ f,
            float* __restrict__ red_out, int mode)
{
    int pix = blockIdx.x;
    int ow = pix & 127, oh = (pix >> 7) & 127, b = pix >> 14;
    int oc = threadIdx.x;

    __shared__ float patch[27];
    __shared__ float red[256];

    if (oc < 27) {
        int ic = oc / 9, r = oc % 9, kh = r / 3, kw = r % 3;
        int ih = oh * 2 - 1 + kh, iw = ow * 2 - 1 + kw;
        float v = 0.0f;
        if (ih >= 0 && ih < 256 && iw >= 0 && iw < 256)
            v = x[((b * 3 + ic) * 256 + ih) * 256 + iw];
        patch[oc] = v;
    }
    __syncthreads();

    float acc = bc[oc];
    const float* w = Wc + oc * 27;
#pragma unroll
    for (int j = 0; j < 27; ++j) acc = fmaf(patch[j], w[j], acc);
    acc = fmaxf(acc, 0.0f);
    f[(b * COUT + oc) * NPIX + oh * OHW + ow] = acc;

    red[oc] = mode ? acc * acc : acc;
    __syncthreads();
    for (int s = 128; s > 0; s >>= 1) {
        if (oc < s) red[oc] += red[oc + s];
        __syncthreads();
    }
    if (oc == 0) red_out[b * NPIX + oh * OHW + ow] = red[0];
}

// ---------------------------------------------------------------------------
// Kernel 2: AdaptiveMaxPool2d(16) argmax on response map -> flat indices.
// grid = B*256 regions, block = 64 (one thread per 8x8 cell element)
// ---------------------------------------------------------------------------
__global__ void fsel_kernel(const float* __restrict__ resp, int* __restrict__ idxA)
{
    int rid = blockIdx.x;
    int b = rid >> 8, ki = (rid >> 4) & 15, kj = rid & 15;
    int t = threadIdx.x;
    int lr = t >> 3, lc = t & 7;
    int row = ki * 8 + lr, col = kj * 8 + lc;
    float v = resp[b * NPIX + row * OHW + col];

    __shared__ float sv[64];
    __shared__ int   si[64];
    sv[t] = v; si[t] = t;           // t == loc = lr*8+lc, matches jnp layout
    __syncthreads();
    for (int s = 32; s > 0; s >>= 1) {
        if (t < s) {
            // argmax, first occurrence on ties (lowest loc wins)
            if (sv[t + s] > sv[t] || (sv[t + s] == sv[t] && si[t + s] < si[t])) {
                sv[t] = sv[t + s]; si[t] = si[t + s];
            }
        }
        __syncthreads();
    }
    if (t == 0) {
        int loc = si[0];
        int r2 = ki * 8 + loc / 8, c2 = kj * 8 + loc % 8;
        idxA[b * K2 + ki * 16 + kj] = r2 * OHW + c2;
    }
}

// ---------------------------------------------------------------------------
// Kernel 3: gather desc[b][c][k] = fA[b][c][idxA[b][k]]
// ---------------------------------------------------------------------------
__global__ void gather_kernel(const float* __restrict__ fA,
                              const int* __restrict__ idxA,
                              float* __restrict__ desc)
{
    int bc_ = blockIdx.x;
    int b = bc_ >> 8, c = bc_ & 255, k = threadIdx.x;
    int n = idxA[b * K2 + k];
    desc[(b * COUT + c) * K2 + k] = fA[(b * COUT + c) * NPIX + n];
}

__global__ void init_kernel(unsigned long long* __restrict__ packed)
{
    int i = blockIdx.x * 256 + threadIdx.x;
    if (i < B_ * K2) packed[i] = 0xFFFFFFFFFFFFFFFFull;
}

// ---------------------------------------------------------------------------
// Kernel 4: fused batched GEMM (desc^T x fB) + running argmin, full f32 via
// V_WMMA_F32_16X16X4_F32.  dist_rel[k,n] = ||b_n||^2 - 2*dot(desc_k, fB_n)
// grid = B * 16 k-tiles * 8 n-splits, block = 256 (8 waves)
// wave handles 16 n-tiles of 16 cols; A tile (16x256) preloaded in VGPRs.
// ---------------------------------------------------------------------------
__device__ __forceinline__ unsigned int fkey(float v) {
    unsigned int u = __float_as_uint(v);
    return (u & 0x80000000u) ? ~u : (u | 0x80000000u);   // monotone float->uint
}

#define SPLITS 8

__global__ void __launch_bounds__(256)
knn_kernel(const float* __restrict__ desc, const float* __restrict__ fB,
           const float* __restrict__ bn2, unsigned long long* __restrict__ packed)
{
    int id = blockIdx.x;
    int sp = id % SPLITS;
    int kt = (id / SPLITS) % 16;
    int b  = id / (SPLITS * 16);

    int lane = threadIdx.x & 31;
    int wave = threadIdx.x >> 5;
    int half = lane >> 4;       // C/D layout: M = r + 8*half, N = lane&15
    int lm   = lane & 15;

    const float* dB   = desc + b * COUT * K2;    // [c][k], row stride 256
    const float* fBb  = fB   + b * COUT * NPIX;  // [c][n], row stride 16384
    const float* bn2b = bn2  + b * NPIX;
    int k0 = kt * 16;

    // A fragments: 16(M) x 4(C) f32 per WMMA; lane L, vgpr r -> M=L%16, C=r+2*(L/16)
    v2f a[64];
#pragma unroll
    for (int cc = 0; cc < 64; ++cc) {
        int c = cc * 4 + half * 2;
        a[cc].x = dB[c * K2 + k0 + lm];
        a[cc].y = dB[(c + 1) * K2 + k0 + lm];
    }

    float minv[8];
    int   mini[8];
#pragma unroll
    for (int r = 0; r < 8; ++r) { minv[r] = 3.4e38f; mini[r] = 0x7FFFFFFF; }

    int nBase = sp * (NPIX / SPLITS);
    for (int t = 0; t < 16; ++t) {
        int n0 = nBase + (t * 8 + wave) * 16;
        v8f acc = {};
#pragma unroll
        for (int cc = 0; cc < 64; ++cc) {
            int c = cc * 4 + half * 2;
            v2f bf;
            bf.x = fBb[c * NPIX + n0 + lm];        // B 4x16: N=L%16, C=r+2*(L/16)
            bf.y = fBb[(c + 1) * NPIX + n0 + lm];
            acc = __builtin_amdgcn_wmma_f32_16x16x4_f32(
                      false, a[cc], false, bf, (short)0, acc, false, false);
        }
        int n = n0 + lm;
        float bn = bn2b[n];
#pragma unroll
        for (int r = 0; r < 8; ++r) {
            float s = fmaf(-2.0f, acc[r], bn);
            if (s < minv[r]) { minv[r] = s; mini[r] = n; }
        }
    }

    // reduce over the 16 lanes that share each M row (wave32, width-16 xor)
#pragma unroll
    for (int r = 0; r < 8; ++r) {
        float v = minv[r];
        int   i = mini[r];
#pragma unroll
        for (int off = 8; off > 0; off >>= 1) {
            float ov = __shfl_xor(v, off, 16);
            int   oi = __shfl_xor(i, off, 16);
            if (ov < v || (ov == v && oi < i)) { v = ov; i = oi; }
        }
        if (lm == 0) {
            int M = r + half * 8;
            unsigned long long key =
                ((unsigned long long)fkey(v) << 32) | (unsigned int)i;
            atomicMin(&packed[b * K2 + k0 + M], key);
        }
    }
}

// ---------------------------------------------------------------------------
// Kernel 5: coordinate deltas + two tiny MLPs (256->128 relu ->1), out [B,2]
// grid = B, block = 128 (one thread per hidden unit)
// ---------------------------------------------------------------------------
__global__ void __launch_bounds__(128)
head_kernel(const int* __restrict__ idxA, const unsigned long long* __restrict__ packed,
            const float* __restrict__ W1r, const float* __restrict__ b1r,
            const float* __restrict__ W2r, const float* __restrict__ b2r,
            const float* __restrict__ W1c, const float* __restrict__ b1c,
            const float* __restrict__ W2c, const float* __restrict__ b2c,
            float* __restrict__ out)
{
    int b = blockIdx.x, h = threadIdx.x;
    __shared__ float drow[K2], dcol[K2], red[HID];

    for (int k = h; k < K2; k += HID) {
        int ia = idxA[b * K2 + k];
        int ib = (int)(packed[b * K2 + k] & 0xFFFFFFFFull);
        int rA = ia / OHW, cA = ia % OHW;
        int rB = ib / OHW, cB = ib % OHW;
        drow[k] = (float)(rB - rA);
        dcol[k] = (float)(cA - cB);
    }
    __syncthreads();

    float hr = b1r[h], hc = b1c[h];
    for (int k = 0; k < K2; ++k) {
        hr = fmaf(drow[k], W1r[k * HID + h], hr);
        hc = fmaf(dcol[k], W1c[k * HID + h], hc);
    }
    hr = fmaxf(hr, 0.0f);
    hc = fmaxf(hc, 0.0f);

    red[h] = hr * W2r[h];
    __syncthreads();
    for (int s = 64; s > 0; s >>= 1) { if (h < s) red[h] += red[h + s]; __syncthreads(); }
    if (h == 0) out[b * 2 + 0] = red[0] + b2r[0];
    __syncthreads();

    red[h] = hc * W2c[h];
    __syncthreads();
    for (int s = 64; s > 0; s >>= 1) { if (h < s) red[h] += red[h + s]; __syncthreads(); }
    if (h == 0) out[b * 2 + 1] = red[0] + b2c[0];
}

// ---------------------------------------------------------------------------
extern "C" void kernel_launch(void* const* d_in, const int* in_sizes, int n_in,
                              void* d_out, int out_size, void* d_ws, size_t ws_size,
                              hipStream_t stream)
{
    const float* xA    = (const float*)d_in[0];
    const float* xB    = (const float*)d_in[1];
    const float* Wconv = (const float*)d_in[2];
    const float* bconv = (const float*)d_in[3];
    const float* W1r   = (const float*)d_in[4];
    const float* b1r   = (const float*)d_in[5];
    const float* W2r   = (const float*)d_in[6];
    const float* b2r   = (const float*)d_in[7];
    const float* W1c   = (const float*)d_in[8];
    const float* b1c   = (const float*)d_in[9];
    const float* W2c   = (const float*)d_in[10];
    const float* b2c   = (const float*)d_in[11];
    float* out = (float*)d_out;

    // workspace layout (bytes, 256-aligned)
    char* ws = (char*)d_ws;
    size_t off = 0;
    auto alloc = [&](size_t bytes) { char* p = ws + off; off = (off + bytes + 255) & ~(size_t)255; return p; };
    float* fA    = (float*)alloc((size_t)B_ * COUT * NPIX * 4);   // 67 MB
    float* fB    = (float*)alloc((size_t)B_ * COUT * NPIX * 4);   // 67 MB
    float* respA = (float*)alloc((size_t)B_ * NPIX * 4);
    float* bn2   = (float*)alloc((size_t)B_ * NPIX * 4);
    float* desc  = (float*)alloc((size_t)B_ * COUT * K2 * 4);
    int*   idxA  = (int*)  alloc((size_t)B_ * K2 * 4);
    unsigned long long* packed = (unsigned long long*)alloc((size_t)B_ * K2 * 8);

    conv_kernel<<<B_ * NPIX, 256, 0, stream>>>(xA, Wconv, bconv, fA, respA, 0);
    conv_kernel<<<B_ * NPIX, 256, 0, stream>>>(xB, Wconv, bconv, fB, bn2, 1);
    fsel_kernel<<<B_ * K2, 64, 0, stream>>>(respA, idxA);
    gather_kernel<<<B_ * COUT, 256, 0, stream>>>(fA, idxA, desc);
    init_kernel<<<(B_ * K2 + 255) / 256, 256, 0, stream>>>(packed);
    knn_kernel<<<B_ * 16 * SPLITS, 256, 0, stream>>>(desc, fB, bn2, packed);
    head_kernel<<<B_, HID, 0, stream>>>(idxA, packed, W1r, b1r, W2r, b2r,
                                        W1c, b1c, W2c, b2c, out);
}